// MambaComplex_57836029608127
// MI455X (gfx1250) — compile-verified
//
#include <hip/hip_runtime.h>
#include <math.h>

// ---------------------------------------------------------------------------
// Problem constants (match the JAX reference)
// ---------------------------------------------------------------------------
#define D_MODEL   768
#define D_STATE   16
#define D_CONV    4
#define D_INNER   1536          // 2 * 768
#define DT_RANK   48
#define BATCH     2
#define SEQLEN    1024
#define NTOK      (BATCH * SEQLEN)          // 2048
#define XP_DIM    (DT_RANK + 4 * D_STATE)   // 112

typedef __attribute__((ext_vector_type(2))) float v2f;
typedef __attribute__((ext_vector_type(8))) float v8f;

#define WAVES_PER_BLOCK 8

// ---------------------------------------------------------------------------
// Generic fp32 WMMA GEMM:  C[M,N] = A[M,K(lda)] * W[N,K(ldw)]^T
// One 16x16 output tile per wave32, K consumed 4 at a time via
// V_WMMA_F32_16X16X4_F32, unrolled x4 per loop iteration.
//
// EPI 0: out[m*N + n] = acc                                  (plain)
// EPI 1: split columns into x / z channel-major buffers      (in_proj)
// EPI 2: out[m*N + n] = softplus(acc + bias[n])              (dt_proj)
// ---------------------------------------------------------------------------
template <int EPI>
__global__ __launch_bounds__(256)
void gemm_wmma_f32(const float* __restrict__ A, const float* __restrict__ W,
                   int M, int N, int K, int lda, int ldw,
                   float* __restrict__ out, float* __restrict__ out2,
                   const float* __restrict__ bias)
{
    const int lane = threadIdx.x & 31;
    const int wave = threadIdx.x >> 5;
    const int tile = blockIdx.x * WAVES_PER_BLOCK + wave;
    const int ntn  = N >> 4;
    const int ntm  = M >> 4;
    if (tile >= ntm * ntn) return;      // wave-uniform: EXEC stays all-1s
    const int tn = tile % ntn;
    const int tm = tile / ntn;

    const int r     = lane & 15;        // row within A tile / col within B tile
    const int khalf = (lane >> 4) << 1; // lanes 0-15 -> K=0,1 ; lanes 16-31 -> K=2,3

    const float* __restrict__ Arow = A + (size_t)(tm * 16 + r) * lda + khalf;
    const float* __restrict__ Wrow = W + (size_t)(tn * 16 + r) * ldw + khalf;

    v8f acc = {};
    for (int k0 = 0; k0 < K; k0 += 16) {
#pragma unroll
        for (int kk = 0; kk < 16; kk += 4) {
            v2f af, bf;
            af[0] = Arow[k0 + kk];
            af[1] = Arow[k0 + kk + 1];
            bf[0] = Wrow[k0 + kk];
            bf[1] = Wrow[k0 + kk + 1];
            acc = __builtin_amdgcn_wmma_f32_16x16x4_f32(
                /*neg_a=*/false, af, /*neg_b=*/false, bf,
                /*c_mod=*/(short)0, acc, /*reuse_a=*/false, /*reuse_b=*/false);
        }
    }

    // C/D layout: VGPR i, lanes 0-15 -> M=i ; lanes 16-31 -> M=i+8 ; N = lane&15
    const int ncol = tn * 16 + r;
    const int mrow0 = tm * 16 + ((lane >> 4) << 3);
#pragma unroll
    for (int i = 0; i < 8; ++i) {
        const int mg = mrow0 + i;
        const float v = acc[i];
        if (EPI == 0) {
            out[(size_t)mg * N + ncol] = v;
        } else if (EPI == 1) {
            // m = token = b*SEQLEN + l ; n = e in [0, 2*D_INNER)
            const int b = mg >> 10;
            const int l = mg & (SEQLEN - 1);
            if (ncol < D_INNER)
                out[((size_t)(b * D_INNER + ncol)) * SEQLEN + l] = v;          // x
            else
                out2[((size_t)(b * D_INNER + (ncol - D_INNER))) * SEQLEN + l] = v; // z
        } else { // EPI == 2 : softplus(acc + bias[n])
            const float t = v + bias[ncol];
            const float sp = (t > 20.0f) ? t : log1pf(expf(t));
            out[(size_t)mg * N + ncol] = sp;
        }
    }
}

// ---------------------------------------------------------------------------
// Depthwise causal conv (width 4) + bias + SiLU.
// Reads x channel-major [B, D_INNER, L]; writes token-major [NTOK, D_INNER].
// ---------------------------------------------------------------------------
__global__ __launch_bounds__(256)
void conv_silu_kernel(const float* __restrict__ xbuf,
                      const float* __restrict__ conv_w,
                      const float* __restrict__ conv_b,
                      float* __restrict__ xc)
{
    const int gid = blockIdx.x * blockDim.x + threadIdx.x;
    if (gid >= BATCH * D_INNER * SEQLEN) return;
    const int l = gid & (SEQLEN - 1);
    const int rest = gid >> 10;
    const int d = rest % D_INNER;
    const int b = rest / D_INNER;

    const float* __restrict__ xrow = xbuf + (size_t)(b * D_INNER + d) * SEQLEN;
    float acc = conv_b[d];
#pragma unroll
    for (int j = 0; j < D_CONV; ++j) {
        const int li = l - (D_CONV - 1) + j;
        if (li >= 0) acc += conv_w[d * D_CONV + j] * xrow[li];
    }
    const float s = acc / (1.0f + expf(-acc));   // SiLU
    xc[((size_t)(b * SEQLEN + l)) * D_INNER + d] = s;
}

// ---------------------------------------------------------------------------
// Complex selective scan. 16 lanes per (b,d) channel, one lane per state n.
// h_{l} = h_{l-1} * exp(delta*A) + delta*x*B_l ;  y_l = 2*Re(sum_n h_n * C_n)
// Then y = (y + D*x) * silu(z). Output token-major [NTOK, D_INNER].
// ---------------------------------------------------------------------------
__global__ __launch_bounds__(256)
void scan_kernel(const float* __restrict__ delta,      // [NTOK, D_INNER]
                 const float* __restrict__ xc,         // [NTOK, D_INNER]
                 const float* __restrict__ xdbl,       // [NTOK, XP_DIM]
                 const float* __restrict__ zbuf,       // [B, D_INNER, L]
                 const float* __restrict__ A_log_real, // [D_INNER, D_STATE]
                 const float* __restrict__ A_imag,     // [D_INNER, D_STATE]
                 const float* __restrict__ Dvec,       // [D_INNER]
                 float* __restrict__ ybuf)             // [NTOK, D_INNER]
{
    const int gid = blockIdx.x * blockDim.x + threadIdx.x;
    if (gid >= BATCH * D_INNER * D_STATE) return;
    const int n  = gid & (D_STATE - 1);
    const int ch = gid >> 4;
    const int d  = ch % D_INNER;
    const int b  = ch / D_INNER;

    const float Are = -expf(A_log_real[d * D_STATE + n]);
    const float Aim = A_imag[d * D_STATE + n];
    const float Dd  = Dvec[d];

    float hre = 0.0f, him = 0.0f;
    for (int l = 0; l < SEQLEN; ++l) {
        const int tok = b * SEQLEN + l;
        const float dl = delta[(size_t)tok * D_INNER + d];
        const float xv = xc[(size_t)tok * D_INNER + d];
        const float* __restrict__ pc = xdbl + (size_t)tok * XP_DIM;
        const float Bre = pc[DT_RANK + 2 * n];
        const float Bim = pc[DT_RANK + 2 * n + 1];
        const float Cre = pc[DT_RANK + 2 * D_STATE + 2 * n];
        const float Cim = pc[DT_RANK + 2 * D_STATE + 2 * n + 1];

        const float e  = expf(dl * Are);
        float sn, cs;
        sincosf(dl * Aim, &sn, &cs);
        const float dAre = e * cs;
        const float dAim = e * sn;
        const float dx   = dl * xv;

        const float nhre = hre * dAre - him * dAim + dx * Bre;
        const float nhim = hre * dAim + him * dAre + dx * Bim;
        hre = nhre; him = nhim;

        float yp = hre * Cre - him * Cim;
#pragma unroll
        for (int off = 8; off >= 1; off >>= 1)
            yp += __shfl_xor(yp, off, 16);

        if (n == 0) {
            const float zz = zbuf[((size_t)(b * D_INNER + d)) * SEQLEN + l];
            const float zs = zz / (1.0f + expf(-zz));
            ybuf[(size_t)tok * D_INNER + d] = (2.0f * yp + Dd * xv) * zs;
        }
    }
}

// ---------------------------------------------------------------------------
// Host-side orchestration
// ---------------------------------------------------------------------------
extern "C" void kernel_launch(void* const* d_in, const int* in_sizes, int n_in,
                              void* d_out, int out_size, void* d_ws, size_t ws_size,
                              hipStream_t stream)
{
    (void)in_sizes; (void)n_in; (void)out_size; (void)ws_size;

    const float* hidden     = (const float*)d_in[0];  // [B, L, D_MODEL]
    const float* in_proj_w  = (const float*)d_in[1];  // [2*D_INNER, D_MODEL]
    const float* conv_w     = (const float*)d_in[2];  // [D_INNER, D_CONV]
    const float* conv_b     = (const float*)d_in[3];  // [D_INNER]
    const float* x_proj_w   = (const float*)d_in[4];  // [XP_DIM, D_INNER]
    const float* dt_proj_w  = (const float*)d_in[5];  // [D_INNER, DT_RANK]
    const float* dt_proj_b  = (const float*)d_in[6];  // [D_INNER]
    const float* A_log_real = (const float*)d_in[7];  // [D_INNER, D_STATE]
    const float* A_imag     = (const float*)d_in[8];  // [D_INNER, D_STATE]
    const float* Dvec       = (const float*)d_in[9];  // [D_INNER]
    const float* out_proj_w = (const float*)d_in[10]; // [D_MODEL, D_INNER]
    float* outp = (float*)d_out;                      // [B, L, D_MODEL]

    // Workspace layout (floats)
    const size_t SZ_CH  = (size_t)BATCH * D_INNER * SEQLEN;  // 3,145,728
    const size_t SZ_XD  = (size_t)NTOK * XP_DIM;             //   229,376
    float* ws    = (float*)d_ws;
    float* xbuf  = ws;                 // x   [B, D_INNER, L]
    float* zbuf  = xbuf  + SZ_CH;      // z   [B, D_INNER, L]
    float* xc    = zbuf  + SZ_CH;      // conv+silu(x)  [NTOK, D_INNER]
    float* xdbl  = xc    + SZ_CH;      // x_dbl [NTOK, XP_DIM]
    float* delta = xdbl  + SZ_XD;      // delta [NTOK, D_INNER]
    float* ybuf  = delta + SZ_CH;      // y     [NTOK, D_INNER]

    const dim3 blk(256);
    auto blocks_for_tiles = [](int m, int n) {
        const int tiles = (m >> 4) * (n >> 4);
        return (tiles + WAVES_PER_BLOCK - 1) / WAVES_PER_BLOCK;
    };

    // 1) in_proj GEMM: [NTOK, D_MODEL] x [2*D_INNER, D_MODEL]^T -> x, z (channel-major)
    gemm_wmma_f32<1><<<blocks_for_tiles(NTOK, 2 * D_INNER), blk, 0, stream>>>(
        hidden, in_proj_w, NTOK, 2 * D_INNER, D_MODEL, D_MODEL, D_MODEL,
        xbuf, zbuf, nullptr);

    // 2) depthwise causal conv + SiLU -> xc [NTOK, D_INNER]
    {
        const int nthr = BATCH * D_INNER * SEQLEN;
        conv_silu_kernel<<<(nthr + 255) / 256, blk, 0, stream>>>(xbuf, conv_w, conv_b, xc);
    }

    // 3) x_proj GEMM: [NTOK, D_INNER] x [XP_DIM, D_INNER]^T -> x_dbl [NTOK, XP_DIM]
    gemm_wmma_f32<0><<<blocks_for_tiles(NTOK, XP_DIM), blk, 0, stream>>>(
        xc, x_proj_w, NTOK, XP_DIM, D_INNER, D_INNER, D_INNER,
        xdbl, nullptr, nullptr);

    // 4) dt_proj GEMM + softplus: dtr = x_dbl[:, :DT_RANK] (lda = XP_DIM)
    gemm_wmma_f32<2><<<blocks_for_tiles(NTOK, D_INNER), blk, 0, stream>>>(
        xdbl, dt_proj_w, NTOK, D_INNER, DT_RANK, XP_DIM, DT_RANK,
        delta, nullptr, dt_proj_b);

    // 5) complex selective scan + D*x skip + SiLU(z) gate -> ybuf [NTOK, D_INNER]
    {
        const int nthr = BATCH * D_INNER * D_STATE;   // 49152
        scan_kernel<<<(nthr + 255) / 256, blk, 0, stream>>>(
            delta, xc, xdbl, zbuf, A_log_real, A_imag, Dvec, ybuf);
    }

    // 6) out_proj GEMM: [NTOK, D_INNER] x [D_MODEL, D_INNER]^T -> out [NTOK, D_MODEL]
    gemm_wmma_f32<0><<<blocks_for_tiles(NTOK, D_MODEL), blk, 0, stream>>>(
        ybuf, out_proj_w, NTOK, D_MODEL, D_INNER, D_INNER, D_INNER,
        outp, nullptr, nullptr);
}